// JaxRateModel_12257836663149
// MI455X (gfx1250) — compile-verified
//
#include <hip/hip_runtime.h>
#include <hip/hip_bf16.h>
#include <stdint.h>

// ---------------------------------------------------------------------------
// Rate-model graph kernel for MI455X (gfx1250).
//   syn[v]  = sum_{e: dst[e]==v} rates[src[e]] * weights[e]
//   out[v]  = (-rates[v] + 10^gain[v] * tanh((is_input? ext : syn) + base)) / tau
//
// Memory-bound: 76.8MB read-once edge stream @ 23.3TB/s ~= 3.4us floor.
// rates / syn are 400KB each -> L2 resident (192MB L2); edge stream is staged
// into LDS with CDNA5 async global->LDS copies (ASYNCcnt), double buffered,
// and marked non-temporal so it does not evict the hot node arrays.
// ---------------------------------------------------------------------------

#define EDGE_BLOCK   256          // threads per block (8 waves)
#define EPT          8            // edges per thread per chunk
#define CHUNK        (EDGE_BLOCK * EPT)   // 2048 edges per chunk

// One 16-byte async copy global -> LDS (GV mode: 64-bit vaddr, no saddr).
// Edge data is single-use: non-temporal so L2 capacity stays with rates/syn.
__device__ __forceinline__ void async_cp16(void* lds_ptr, const void* gptr) {
    // Low 32 bits of a generic shared pointer == workgroup-relative LDS byte
    // address (flat-aperture layout, ISA 10.2).
    uint32_t l = (uint32_t)(uintptr_t)lds_ptr;
    uint64_t g = (uint64_t)(uintptr_t)gptr;
    asm volatile("global_load_async_to_lds_b128 %0, %1, off th:TH_LOAD_NT"
                 :
                 : "v"(l), "v"(g)
                 : "memory");
}

__device__ __forceinline__ void wait_async_le(void) {}
__device__ __forceinline__ void wait_async_le6() {
    asm volatile("s_wait_asynccnt 0x6" ::: "memory");
}
__device__ __forceinline__ void wait_async_le0() {
    asm volatile("s_wait_asynccnt 0x0" ::: "memory");
}

__global__ void __launch_bounds__(EDGE_BLOCK)
rate_zero_kernel(float* __restrict__ syn, int n) {
    int i = blockIdx.x * blockDim.x + threadIdx.x;
    if (i < n) syn[i] = 0.0f;
}

__global__ void __launch_bounds__(EDGE_BLOCK)
rate_edge_kernel(const float* __restrict__ rates,
                 const float* __restrict__ weights,
                 const int*   __restrict__ src,
                 const int*   __restrict__ dst,
                 float*       __restrict__ syn,
                 long long E, int nChunks) {
    __shared__ __align__(16) float sW[2][CHUNK];
    __shared__ __align__(16) int   sS[2][CHUNK];
    __shared__ __align__(16) int   sD[2][CHUNK];

    const int tid  = threadIdx.x;
    const int eBeg = tid * EPT;           // this lane's first element in the chunk

    int c = blockIdx.x;
    if (c >= nChunks) return;
    const int stride = gridDim.x;

    auto chunk_full = [&](int ch) -> bool {
        return ((long long)ch + 1) * CHUNK <= E;
    };
    // 6 async b128 copies per lane per chunk: 2 each for weights/src/dst.
    auto issue = [&](int ch, int buf) {
        size_t base = (size_t)ch * CHUNK + (size_t)eBeg;
        async_cp16(&sW[buf][eBeg],     weights + base);
        async_cp16(&sW[buf][eBeg + 4], weights + base + 4);
        async_cp16(&sS[buf][eBeg],     src + base);
        async_cp16(&sS[buf][eBeg + 4], src + base + 4);
        async_cp16(&sD[buf][eBeg],     dst + base);
        async_cp16(&sD[buf][eBeg + 4], dst + base + 4);
    };

    int cur = 0;
    if (chunk_full(c)) issue(c, cur);

    for (;;) {
        const int  nxt     = c + stride;
        const bool hasNext = nxt < nChunks;
        const bool nxtFull = hasNext && chunk_full(nxt);
        if (nxtFull) issue(nxt, cur ^ 1);

        if (chunk_full(c)) {
            // Drain this chunk's 6 async loads (in-order completion), leaving
            // the next chunk's 6 (if issued) in flight.
            if (nxtFull) wait_async_le6();
            else         wait_async_le0();

            #pragma unroll
            for (int h = 0; h < 2; ++h) {
                const int o = eBeg + h * 4;
                const float4 w = *(const float4*)&sW[cur][o];
                const int4   s = *(const int4*)  &sS[cur][o];
                const int4   d = *(const int4*)  &sD[cur][o];
                atomicAdd(&syn[d.x], rates[s.x] * w.x);
                atomicAdd(&syn[d.y], rates[s.y] * w.y);
                atomicAdd(&syn[d.z], rates[s.z] * w.z);
                atomicAdd(&syn[d.w], rates[s.w] * w.w);
            }
        } else {
            // Tail chunk: bounds-checked scalar path straight from global.
            long long e0 = (long long)c * CHUNK + eBeg;
            #pragma unroll
            for (int k = 0; k < EPT; ++k) {
                long long e = e0 + k;
                if (e < E) atomicAdd(&syn[dst[e]], rates[src[e]] * weights[e]);
            }
        }

        if (!hasNext) break;
        c = nxt;
        cur ^= 1;
    }
}

__global__ void __launch_bounds__(256)
rate_node_kernel(const float* __restrict__ rates,
                 const float* __restrict__ gain,
                 const float* __restrict__ tau,
                 const float* __restrict__ baseline,
                 const float* __restrict__ ext,
                 const unsigned char* __restrict__ is_input,
                 const float* __restrict__ syn,
                 float* __restrict__ out, int n) {
    int i = blockIdx.x * blockDim.x + threadIdx.x;
    if (i >= n) return;
    float total = (is_input[i] ? ext[i] : syn[i]) + baseline[i];
    float act   = tanhf(total);
    float g     = __expf(gain[i] * 2.302585092994046f);   // 10^gain
    out[i] = (g * act - rates[i]) / tau[i];
}

extern "C" void kernel_launch(void* const* d_in, const int* in_sizes, int n_in,
                              void* d_out, int out_size, void* d_ws, size_t ws_size,
                              hipStream_t stream) {
    const float*         rates    = (const float*)d_in[0];
    const float*         weights  = (const float*)d_in[1];
    const float*         gain     = (const float*)d_in[2];
    const float*         tau      = (const float*)d_in[3];
    const float*         baseline = (const float*)d_in[4];
    const float*         ext      = (const float*)d_in[5];
    const int*           src      = (const int*)d_in[6];
    const int*           dst      = (const int*)d_in[7];
    const unsigned char* is_input = (const unsigned char*)d_in[8];

    const int       N = in_sizes[0];
    const long long E = (long long)in_sizes[1];

    float* syn = (float*)d_ws;          // N floats of scratch
    float* out = (float*)d_out;

    const int nodeBlocks = (N + 255) / 256;
    rate_zero_kernel<<<nodeBlocks, EDGE_BLOCK, 0, stream>>>(syn, N);

    const int nChunks = (int)((E + CHUNK - 1) / CHUNK);
    int grid = nChunks < 1024 ? nChunks : 1024;   // persistent blocks
    if (grid < 1) grid = 1;
    rate_edge_kernel<<<grid, EDGE_BLOCK, 0, stream>>>(rates, weights, src, dst,
                                                      syn, E, nChunks);

    rate_node_kernel<<<nodeBlocks, 256, 0, stream>>>(rates, gain, tau, baseline,
                                                     ext, is_input, syn, out, N);
}